// PrepareEnvGCN_26774826123929
// MI455X (gfx1250) — compile-verified
//
#include <hip/hip_runtime.h>
#include <hip/hip_bf16.h>

typedef __attribute__((ext_vector_type(2))) float v2f;
typedef __attribute__((ext_vector_type(8))) float v8f;

#define N_NODES  250000
#define N_EDGES  4000000
#define N_GRAPHS 512
#define N_TILES  (N_NODES / 16)   // 15625, exact
#define NEG_SLOPE 0.01f

// ---------------------------------------------------------------- zero fill
__global__ void zero_f32(float* __restrict__ p, long long n) {
  long long i = (long long)blockIdx.x * blockDim.x + threadIdx.x;
  long long stride = (long long)gridDim.x * blockDim.x;
  for (; i < n; i += stride) p[i] = 0.0f;
}

// -------------------------------------- pad x [N,13] -> xpad [N,16] (zero tail)
__global__ void pad_x(const float* __restrict__ x, float* __restrict__ xpad,
                      int n_nodes) {
  int n = blockIdx.x * blockDim.x + threadIdx.x;
  if (n >= n_nodes) return;
  const float* xr = x + (long long)n * 13;
  float r[16];
#pragma unroll
  for (int f = 0; f < 13; ++f) r[f] = xr[f];
  r[13] = r[14] = r[15] = 0.0f;
  float4* o = (float4*)(xpad + (long long)n * 16);
  o[0] = make_float4(r[0], r[1], r[2], r[3]);
  o[1] = make_float4(r[4], r[5], r[6], r[7]);
  o[2] = make_float4(r[8], r[9], r[10], r[11]);
  o[3] = make_float4(r[12], r[13], r[14], r[15]);
}

// ------------------------------------------------------- per-edge in-degree
__global__ void count_edges(const int* __restrict__ dst,
                            float* __restrict__ cnt, int n_edges) {
  int e = blockIdx.x * blockDim.x + threadIdx.x;
  if (e >= n_edges) return;
  atomicAdd(&cnt[dst[e]], 1.0f);
}

// -------------------------- edge scatter: msg[dst][0:D] += h[src][0:D], stride S
template <int D, int S>
__global__ void scatter_edges(const float* __restrict__ h,
                              const int* __restrict__ src,
                              const int* __restrict__ dst,
                              float* __restrict__ msg, int n_edges) {
  int e = blockIdx.x * blockDim.x + threadIdx.x;
  if (e >= n_edges) return;
  int s = src[e];
  int t = dst[e];
  const float4* hs = (const float4*)(h + (long long)s * S);
  float* md = msg + (long long)t * S;
#pragma unroll
  for (int q = 0; q < D / 4; ++q) {
    float4 v = hs[q];
    atomicAdd(&md[q * 4 + 0], v.x);
    atomicAdd(&md[q * 4 + 1], v.y);
    atomicAdd(&md[q * 4 + 2], v.z);
    atomicAdd(&md[q * 4 + 3], v.w);
  }
}

// ------------------------------------------------- WMMA node update
// One wave = one 16-node tile:
//   out = act( (msg/max(cnt,1)) @ Wl + b + x @ Wr )
// f32 WMMA 16x16x4: A 16x4 (v2f/lane), B 4x16 (v2f/lane), C/D 16x16 (v8f/lane).
// Weights staged in LDS transposed+zero-padded so every load is unconditional.
template <int D_IN, int D_OUT, int XS, int MS, int OS, bool RELU>
__global__ void sage_node_update(const float* __restrict__ xin,
                                 const float* __restrict__ msg,
                                 const float* __restrict__ cnt,
                                 const float* __restrict__ Wl,
                                 const float* __restrict__ bias,
                                 const float* __restrict__ Wr,
                                 float* __restrict__ out, int n_tiles) {
  __shared__ float sWlT[16 * 16];  // [n][k], zero-padded
  __shared__ float sWrT[16 * 16];
  __shared__ float sB[16];

  // 256 threads cover the 16x16 stage exactly
  {
    int t = threadIdx.x;
    int n = t >> 4, k = t & 15;
    bool in = (k < D_IN) & (n < D_OUT);
    sWlT[n * 16 + k] = in ? Wl[k * D_OUT + n] : 0.0f;
    sWrT[n * 16 + k] = in ? Wr[k * D_OUT + n] : 0.0f;
    if (t < 16) sB[t] = (t < D_OUT) ? bias[t] : 0.0f;
  }
  __syncthreads();

  int wave = (blockIdx.x * blockDim.x + threadIdx.x) >> 5;
  int lane = threadIdx.x & 31;
  if (wave >= n_tiles) return;  // wave-uniform; EXEC all-1s below

  int half = lane >> 4;   // selects K pair {k0,k0+1} vs {k0+2,k0+3}
  int l16 = lane & 15;
  int node0 = wave * 16;
  int node = node0 + l16;  // A-matrix row for this lane (N_NODES % 16 == 0)

  float inv = 1.0f / fmaxf(cnt[node], 1.0f);

  float bn = sB[l16];
  v8f acc;
#pragma unroll
  for (int r = 0; r < 8; ++r) acc[r] = bn;

  const float* mrow = msg + (long long)node * MS;
  const float* xrow = xin + (long long)node * XS;
  const float* wlrow = sWlT + l16 * 16;
  const float* wrrow = sWrT + l16 * 16;

  constexpr int KCH = (D_IN + 3) / 4;  // K chunks of 4 (padded region is zero)
#pragma unroll
  for (int c = 0; c < KCH; ++c) {
    int ka = c * 4 + 2 * half;  // even -> 8B-aligned v2f loads

    v2f a = *(const v2f*)(mrow + ka);  // msg padded with zeros beyond D_IN
    a.x *= inv;
    a.y *= inv;
    v2f b = *(const v2f*)(wlrow + ka);
    acc = __builtin_amdgcn_wmma_f32_16x16x4_f32(false, a, false, b,
                                                (short)0, acc, false, false);

    v2f a2 = *(const v2f*)(xrow + ka);  // xpad/h buffers padded likewise
    v2f b2 = *(const v2f*)(wrrow + ka);
    acc = __builtin_amdgcn_wmma_f32_16x16x4_f32(false, a2, false, b2,
                                                (short)0, acc, false, false);
  }

  // D layout: VGPR r -> row M = r + 8*half, col N = l16
  if (l16 < D_OUT) {
#pragma unroll
    for (int r = 0; r < 8; ++r) {
      float v = acc[r];
      if (RELU) v = (v >= 0.0f) ? v : v * NEG_SLOPE;
      out[(long long)(node0 + r + 8 * half) * OS + l16] = v;
    }
  }
}

// --------------------------------------------------------------- graph pool
__global__ void pool_nodes(const float* __restrict__ h3,
                           const int* __restrict__ batch,
                           float* __restrict__ add_pool,
                           float* __restrict__ gcnt, int n_nodes) {
  int n = blockIdx.x * blockDim.x + threadIdx.x;
  if (n >= n_nodes) return;
  int g = batch[n];
  float4 v = *(const float4*)(h3 + (long long)n * 4);
  atomicAdd(&add_pool[g * 4 + 0], v.x);
  atomicAdd(&add_pool[g * 4 + 1], v.y);
  atomicAdd(&add_pool[g * 4 + 2], v.z);
  atomicAdd(&add_pool[g * 4 + 3], v.w);
  atomicAdd(&gcnt[g], 1.0f);
}

// ------------------------------------------------- readout: [mean|add]@Wc + bc
__global__ void readout(const float* __restrict__ add_pool,
                        const float* __restrict__ gcnt,
                        const float* __restrict__ Wc,
                        const float* __restrict__ bc,
                        float* __restrict__ out, int n_graphs) {
  int g = blockIdx.x * blockDim.x + threadIdx.x;
  if (g >= n_graphs) return;
  float inv = 1.0f / fmaxf(gcnt[g], 1.0f);
  float s = bc[0];
#pragma unroll
  for (int f = 0; f < 4; ++f) {
    float ap = add_pool[g * 4 + f];
    s += ap * inv * Wc[f];  // mean_pool half of concat
    s += ap * Wc[4 + f];    // add_pool half
  }
  out[g] = s;
}

extern "C" void kernel_launch(void* const* d_in, const int* in_sizes, int n_in,
                              void* d_out, int out_size, void* d_ws, size_t ws_size,
                              hipStream_t stream) {
  const float* x   = (const float*)d_in[0];
  const int* ei    = (const int*)d_in[1];  // [2, N_EDGES]
  const int* src   = ei;
  const int* dst   = ei + N_EDGES;
  const int* batch = (const int*)d_in[2];
  const float* W1l = (const float*)d_in[3];
  const float* b1  = (const float*)d_in[4];
  const float* W1r = (const float*)d_in[5];
  const float* W2l = (const float*)d_in[6];
  const float* b2  = (const float*)d_in[7];
  const float* W2r = (const float*)d_in[8];
  const float* W3l = (const float*)d_in[9];
  const float* b3  = (const float*)d_in[10];
  const float* W3r = (const float*)d_in[11];
  const float* Wc  = (const float*)d_in[12];
  const float* bc  = (const float*)d_in[13];
  float* out = (float*)d_out;

  // Workspace layout (floats); h2/h3 alias xpad (dead after layer 1).
  float* ws   = (float*)d_ws;
  float* cnt  = ws;                              // 250,000
  float* msg  = cnt + N_NODES;                   // 4,000,000 (stride 16 padded)
  float* xpad = msg + (long long)N_NODES * 16;   // 4,000,000
  float* h1   = xpad + (long long)N_NODES * 16;  // 4,000,000
  float* h2   = xpad;                            // 2,000,000 (alias)
  float* h3   = xpad + (long long)N_NODES * 8;   // 1,000,000 (alias)
  float* add_pool = h1 + (long long)N_NODES * 16;  // 2048
  float* gcnt = add_pool + N_GRAPHS * 4;           // 512

  const int TB = 256;
  const int edge_blocks = (N_EDGES + TB - 1) / TB;
  const int node_blocks = (N_NODES + TB - 1) / TB;
  const int upd_blocks = (N_TILES * 32 + TB - 1) / TB;  // 8 tiles / block

  // zero accumulators (cnt+msg contiguous) and pools
  zero_f32<<<2048, TB, 0, stream>>>(cnt, (long long)N_NODES + (long long)N_NODES * 16);
  zero_f32<<<8, TB, 0, stream>>>(add_pool, N_GRAPHS * 4 + N_GRAPHS);

  pad_x<<<node_blocks, TB, 0, stream>>>(x, xpad, N_NODES);
  count_edges<<<edge_blocks, TB, 0, stream>>>(dst, cnt, N_EDGES);

  // ---- layer 1: 13 -> 16 (xpad stride 16, zero-padded -> clean vec loads)
  scatter_edges<16, 16><<<edge_blocks, TB, 0, stream>>>(xpad, src, dst, msg, N_EDGES);
  sage_node_update<13, 16, 16, 16, 16, true>
      <<<upd_blocks, TB, 0, stream>>>(xpad, msg, cnt, W1l, b1, W1r, h1, N_TILES);

  // ---- layer 2: 16 -> 8
  zero_f32<<<2048, TB, 0, stream>>>(msg, (long long)N_NODES * 16);
  scatter_edges<16, 16><<<edge_blocks, TB, 0, stream>>>(h1, src, dst, msg, N_EDGES);
  sage_node_update<16, 8, 16, 16, 8, true>
      <<<upd_blocks, TB, 0, stream>>>(h1, msg, cnt, W2l, b2, W2r, h2, N_TILES);

  // ---- layer 3: 8 -> 4 (no activation)
  zero_f32<<<2048, TB, 0, stream>>>(msg, (long long)N_NODES * 8);
  scatter_edges<8, 8><<<edge_blocks, TB, 0, stream>>>(h2, src, dst, msg, N_EDGES);
  sage_node_update<8, 4, 8, 8, 4, false>
      <<<upd_blocks, TB, 0, stream>>>(h2, msg, cnt, W3l, b3, W3r, h3, N_TILES);

  // ---- pooling + readout
  pool_nodes<<<node_blocks, TB, 0, stream>>>(h3, batch, add_pool, gcnt, N_NODES);
  readout<<<(N_GRAPHS + TB - 1) / TB, TB, 0, stream>>>(add_pool, gcnt, Wc, bc, out, N_GRAPHS);
}